// CausalityEmbedding_70531952935038
// MI455X (gfx1250) — compile-verified
//
#include <hip/hip_runtime.h>
#include <hip/hip_bf16.h>
#include <math.h>

// ---------------- problem constants ----------------
#define NN   1024      // N features
#define HH   256       // H hidden
#define FF   1280      // N + H
#define EE   64        // E embed
#define HD   64        // head dim
#define BB   8192      // batch

typedef float v2f __attribute__((ext_vector_type(2)));
typedef float v8f __attribute__((ext_vector_type(8)));

// ---------------- device math helpers ----------------
__device__ __forceinline__ float fexp2(float x) {
#if defined(__HIP_DEVICE_COMPILE__)
  return __builtin_amdgcn_exp2f(x);        // v_exp_f32
#else
  return exp2f(x);
#endif
}
__device__ __forceinline__ float frcp(float x) {
#if defined(__HIP_DEVICE_COMPILE__)
  return __builtin_amdgcn_rcpf(x);         // v_rcp_f32
#else
  return 1.0f / x;
#endif
}
// tanh(x) = 1 - 2/(exp(2x)+1), branch-free, 1 trans + 1 rcp
__device__ __forceinline__ float fast_tanh(float x) {
  float e = fexp2(x * 2.885390082f);       // exp(2x) = 2^(2x*log2 e)
  return 1.0f - 2.0f * frcp(e + 1.0f);
}
__device__ __forceinline__ float fexp(float x) {
  return fexp2(x * 1.442695041f);
}

// ---------------- WMMA wrapper: D = A(16x4,f32) * B(4x16,f32) + C ----------
// A frag: lane holds A[m][2*kh+{0,1}]   (m = lane&15, kh = lane>>4)
// B frag: lane holds B[2*kh+{0,1}][n]   (n = lane&15)
// C/D   : vgpr r, lane l -> M = r + 8*(l>=16), N = l&15
#if defined(__HIP_DEVICE_COMPILE__) && __has_builtin(__builtin_amdgcn_wmma_f32_16x16x4_f32)
#define HAVE_WMMA_F32X4 1
#endif

__device__ __forceinline__ v8f wmma4(v2f a, v2f b, v8f c) {
#if defined(HAVE_WMMA_F32X4)
  return __builtin_amdgcn_wmma_f32_16x16x4_f32(
      /*neg_a=*/false, a, /*neg_b=*/false, b,
      /*c_mod=*/(short)0, c, /*reuse_a=*/false, /*reuse_b=*/false);
#else
  c[0] += a[0] * b[0];   // placeholder so host pass / missing-builtin still compiles
  return c;
#endif
}

// ==========================================================================
// K1: a = feat_emb @ W1 ; bp = full @ W2 + b_w ; fullT = full^T (64 x 1280)
// grid: 1280 blocks x 64 threads (block j = one row of `full`)
// ==========================================================================
__global__ void k1_prep(const float* __restrict__ feat_emb,
                        const float* __restrict__ hid_emb,
                        const float* __restrict__ W_w,
                        const float* __restrict__ b_w,
                        float* __restrict__ a_out,
                        float* __restrict__ bp_out,
                        float* __restrict__ fullT) {
  const int j = blockIdx.x;
  const int t = threadIdx.x;          // output column 0..63
  __shared__ float row[EE];
  const float* src = (j < NN) ? (feat_emb + (size_t)j * EE)
                              : (hid_emb + (size_t)(j - NN) * EE);
  row[t] = src[t];
  __syncthreads();

  fullT[(size_t)t * FF + j] = row[t];

  float acc2 = 0.0f, acc1 = 0.0f;
#pragma unroll 8
  for (int k = 0; k < EE; ++k) {
    const float fv = row[k];
    acc2 = fmaf(fv, W_w[(size_t)(EE + k) * HD + t], acc2);   // W2
    acc1 = fmaf(fv, W_w[(size_t)k * HD + t], acc1);          // W1
  }
  bp_out[(size_t)j * HD + t] = acc2 + b_w[t];
  if (j < NN) a_out[(size_t)j * HD + t] = acc1;
}

// ==========================================================================
// K2: attn[i][j] = exp(W_u . tanh(a_i + bp_j)) * mask, row-normalized
// grid: 1024 blocks x 256 threads (block = one query row i; 5 j's per thread)
// ==========================================================================
__global__ void k2_attn(const float* __restrict__ a_in,
                        const float* __restrict__ bp_in,
                        const float* __restrict__ W_u,
                        const unsigned char* __restrict__ mask,
                        float* __restrict__ attn) {
  const int i   = blockIdx.x;
  const int tid = threadIdx.x;
  __shared__ float sA[HD];
  __shared__ float sWu[HD];
  __shared__ float red[256];
  if (tid < HD) {
    sA[tid]  = a_in[(size_t)i * HD + tid];
    sWu[tid] = W_u[tid];
  }
  __syncthreads();

  float ev[5];
  float psum = 0.0f;
#pragma unroll
  for (int jj = 0; jj < 5; ++jj) {
    const int j = tid + jj * 256;
    const float4* bj4 = (const float4*)(bp_in + (size_t)j * HD);
    float s = 0.0f;
#pragma unroll
    for (int k4 = 0; k4 < HD / 4; ++k4) {
      const float4 b4 = bj4[k4];
      s = fmaf(fast_tanh(sA[4 * k4 + 0] + b4.x), sWu[4 * k4 + 0], s);
      s = fmaf(fast_tanh(sA[4 * k4 + 1] + b4.y), sWu[4 * k4 + 1], s);
      s = fmaf(fast_tanh(sA[4 * k4 + 2] + b4.z), sWu[4 * k4 + 2], s);
      s = fmaf(fast_tanh(sA[4 * k4 + 3] + b4.w), sWu[4 * k4 + 3], s);
    }
    const float e = mask[(size_t)i * FF + j] ? fexp(s) : 0.0f;
    ev[jj] = e;
    psum += e;
  }

  // block reduction of row sum
  red[tid] = psum;
  __syncthreads();
#pragma unroll
  for (int off = 128; off > 0; off >>= 1) {
    if (tid < off) red[tid] += red[tid + off];
    __syncthreads();
  }
  float total = red[0];
  total = (total == 0.0f) ? 1.0f : total;
  const float inv = frcp(total);

#pragma unroll
  for (int jj = 0; jj < 5; ++jj) {
    const int j = tid + jj * 256;
    attn[(size_t)i * FF + j] = ev[jj] * inv;
  }
}

// ==========================================================================
// K3: contextT(64 x 1024) = (attn(1024x1280) @ full(1280x64))^T  via f32 WMMA
// one wave = 16(i) x 64(n) tile; grid: 16 blocks x 128 threads (64 waves)
// ==========================================================================
__global__ void k3_ctxT_wmma(const float* __restrict__ attn,
                             const float* __restrict__ fullT,
                             float* __restrict__ ctxT) {
  const int lane = threadIdx.x & 31;
  const int wid  = threadIdx.x >> 5;
  const int i0   = (blockIdx.x * 4 + wid) * 16;      // 0..1008
  const int m    = lane & 15;
  const int kh   = lane >> 4;

  const float* arow = attn  + (size_t)(i0 + m) * FF + 2 * kh;
  const float* b0   = fullT + (size_t)(0 * 16 + m) * FF + 2 * kh;
  const float* b1   = fullT + (size_t)(1 * 16 + m) * FF + 2 * kh;
  const float* b2   = fullT + (size_t)(2 * 16 + m) * FF + 2 * kh;
  const float* b3   = fullT + (size_t)(3 * 16 + m) * FF + 2 * kh;

  v8f c0 = {}, c1 = {}, c2 = {}, c3 = {};
  for (int k = 0; k < FF; k += 4) {
    const v2f af = *(const v2f*)(arow + k);
    c0 = wmma4(af, *(const v2f*)(b0 + k), c0);
    c1 = wmma4(af, *(const v2f*)(b1 + k), c1);
    c2 = wmma4(af, *(const v2f*)(b2 + k), c2);
    c3 = wmma4(af, *(const v2f*)(b3 + k), c3);
  }

#pragma unroll
  for (int r = 0; r < 8; ++r) {
    const int ig = i0 + r + 8 * kh;
    ctxT[(size_t)(0 * 16 + m) * NN + ig] = c0[r];
    ctxT[(size_t)(1 * 16 + m) * NN + ig] = c1[r];
    ctxT[(size_t)(2 * 16 + m) * NN + ig] = c2[r];
    ctxT[(size_t)(3 * 16 + m) * NN + ig] = c3[r];
  }
}

// ==========================================================================
// K4: out(8192x64) = values(8192x1024) @ context(1024x64)  via f32 WMMA
// one wave = 16(row) x 64(n) tile; grid: 128 blocks x 128 threads (512 waves)
// ==========================================================================
__global__ void k4_out_wmma(const float* __restrict__ values,
                            const float* __restrict__ ctxT,
                            float* __restrict__ out) {
  const int lane = threadIdx.x & 31;
  const int wid  = threadIdx.x >> 5;
  const int row0 = (blockIdx.x * 4 + wid) * 16;      // 0..8176
  const int m    = lane & 15;
  const int kh   = lane >> 4;

  const float* arow = values + (size_t)(row0 + m) * NN + 2 * kh;
  const float* b0   = ctxT + (size_t)(0 * 16 + m) * NN + 2 * kh;
  const float* b1   = ctxT + (size_t)(1 * 16 + m) * NN + 2 * kh;
  const float* b2   = ctxT + (size_t)(2 * 16 + m) * NN + 2 * kh;
  const float* b3   = ctxT + (size_t)(3 * 16 + m) * NN + 2 * kh;

  v8f c0 = {}, c1 = {}, c2 = {}, c3 = {};
  for (int k = 0; k < NN; k += 4) {
    const v2f af = *(const v2f*)(arow + k);
    c0 = wmma4(af, *(const v2f*)(b0 + k), c0);
    c1 = wmma4(af, *(const v2f*)(b1 + k), c1);
    c2 = wmma4(af, *(const v2f*)(b2 + k), c2);
    c3 = wmma4(af, *(const v2f*)(b3 + k), c3);
  }

#pragma unroll
  for (int r = 0; r < 8; ++r) {
    const size_t rg = (size_t)(row0 + r + 8 * kh) * HD;
    out[rg + 0 * 16 + m] = c0[r];
    out[rg + 1 * 16 + m] = c1[r];
    out[rg + 2 * 16 + m] = c2[r];
    out[rg + 3 * 16 + m] = c3[r];
  }
}

// ==========================================================================
extern "C" void kernel_launch(void* const* d_in, const int* in_sizes, int n_in,
                              void* d_out, int out_size, void* d_ws, size_t ws_size,
                              hipStream_t stream) {
  const float*         values   = (const float*)d_in[0];          // (8192,1024)
  const float*         feat_emb = (const float*)d_in[1];          // (1024,64)
  const float*         hid_emb  = (const float*)d_in[2];          // (256,64)
  const float*         W_w      = (const float*)d_in[3];          // (128,64)
  const float*         b_w      = (const float*)d_in[4];          // (64,)
  const float*         W_u      = (const float*)d_in[5];          // (64,1)
  const unsigned char* mask     = (const unsigned char*)d_in[6];  // (1024,1280) bool
  float*               out      = (float*)d_out;                  // (8192,64)

  // workspace layout (floats); total 1,605,632 f = 6.4 MB
  float* ws    = (float*)d_ws;
  float* a     = ws;                                  //  65536
  float* bp    = ws +   65536;                        //  81920
  float* fullT = ws +  147456;                        //  81920
  float* attn  = ws +  229376;                        // 1310720
  float* ctxT  = ws + 1540096;                        //  65536

  k1_prep<<<dim3(FF), dim3(EE), 0, stream>>>(feat_emb, hid_emb, W_w, b_w,
                                             a, bp, fullT);
  k2_attn<<<dim3(NN), dim3(256), 0, stream>>>(a, bp, W_u, mask, attn);
  k3_ctxT_wmma<<<dim3(NN / 16 / 4), dim3(128), 0, stream>>>(attn, fullT, ctxT);
  k4_out_wmma<<<dim3(BB / 16 / 4), dim3(128), 0, stream>>>(values, ctxT, out);
}